// GCNwithAttention_67242007986863
// MI455X (gfx1250) — compile-verified
//
#include <hip/hip_runtime.h>
#include <hip/hip_bf16.h>

typedef __attribute__((ext_vector_type(2))) float v2f;
typedef __attribute__((ext_vector_type(8))) float v8f;
typedef __attribute__((ext_vector_type(4))) int   v4i;

// ---------------------------------------------------------------------------
// Async global->LDS staging helpers (gfx1250 GLOBAL_LOAD_ASYNC_TO_LDS_B128,
// tracked by ASYNCcnt). Builtin prototype (from compiler diagnostic):
//   void __builtin_amdgcn_global_load_async_to_lds_b128(
//        global v4i* src, lds v4i* dst, imm int offset, imm int cpol)
// Guarded so the file compiles on toolchains without the builtins.
// ---------------------------------------------------------------------------
#if defined(__gfx1250__) && __has_builtin(__builtin_amdgcn_global_load_async_to_lds_b128)
typedef __attribute__((address_space(1))) v4i* gv4i_p;
typedef __attribute__((address_space(3))) v4i* lv4i_p;
#endif

__device__ __forceinline__ void stage_f4(const float* __restrict__ g, float* l) {
#if defined(__gfx1250__) && __has_builtin(__builtin_amdgcn_global_load_async_to_lds_b128)
    v4i* gp = (v4i*)g;   // generic, const dropped
    v4i* lp = (v4i*)l;   // generic
    __builtin_amdgcn_global_load_async_to_lds_b128((gv4i_p)gp, (lv4i_p)lp, 0, 0);
#else
    *(float4*)l = *(const float4*)g;
#endif
}

__device__ __forceinline__ void stage_wait() {
#if defined(__gfx1250__) && __has_builtin(__builtin_amdgcn_s_wait_asynccnt)
    __builtin_amdgcn_s_wait_asynccnt(0);
#endif
}

// ---------------------------------------------------------------------------
// Utility kernels
// ---------------------------------------------------------------------------
__global__ void k_zero(float* __restrict__ p, int n) {
    int i = blockIdx.x * blockDim.x + threadIdx.x;
    if (i < n) p[i] = 0.0f;
}

__global__ void k_degree(const long long* __restrict__ dst, float* __restrict__ deg, int nE) {
    int e = blockIdx.x * blockDim.x + threadIdx.x;
    if (e < nE) atomicAdd(&deg[(int)dst[e]], 1.0f);
}

__global__ void k_dinv(const float* __restrict__ deg, float* __restrict__ dinv, int n) {
    int i = blockIdx.x * blockDim.x + threadIdx.x;
    if (i < n) dinv[i] = rsqrtf(deg[i] + 1.0f);
}

__global__ void k_relu(float* __restrict__ p, int n) {
    int i = blockIdx.x * blockDim.x + threadIdx.x;
    if (i < n) p[i] = fmaxf(p[i], 0.0f);
}

// ---------------------------------------------------------------------------
// Node-feature GEMM:  H = X @ W  (X: n x K row-major, W: K x 64 row-major)
// Fused epilogue: H written, and ACC = H * dinv[row]^2 + b (self-loop + bias).
// One wave computes a 16x64 tile via V_WMMA_F32_16X16X4_F32; 8 waves/block
// -> 128 rows per block. W staged in LDS via async global->LDS loads.
// ---------------------------------------------------------------------------
template <int K>
__global__ __launch_bounds__(256) void k_gemm_node(
    const float* __restrict__ X, const float* __restrict__ Wg,
    const float* __restrict__ bias, const float* __restrict__ dinv,
    float* __restrict__ H, float* __restrict__ ACC, int n)
{
    __shared__ __align__(16) float lw[K * 64];
    for (int i = threadIdx.x * 4; i < K * 64; i += 256 * 4)
        stage_f4(Wg + i, lw + i);
    stage_wait();
    __syncthreads();

    const int lane = threadIdx.x & 31;
    const int wave = threadIdx.x >> 5;
    const bool hi  = lane >= 16;
    const int l16  = lane & 15;

    // Row used for A-fragment loads (clamped so EXEC stays all-ones for WMMA)
    int arow = blockIdx.x * 128 + wave * 16 + l16;
    if (arow > n - 1) arow = n - 1;

    v8f acc[4] = {};

    for (int k = 0; k < K; k += 4) {
        const int kb = k + (hi ? 2 : 0);
        v2f a;
        a.x = X[(size_t)arow * K + kb];
        a.y = X[(size_t)arow * K + kb + 1];
#pragma unroll
        for (int t = 0; t < 4; ++t) {
            const int col = l16 + t * 16;
            v2f bf;
            bf.x = lw[kb * 64 + col];
            bf.y = lw[(kb + 1) * 64 + col];
            acc[t] = __builtin_amdgcn_wmma_f32_16x16x4_f32(
                false, a, false, bf, (short)0, acc[t], false, false);
        }
    }

    // C/D layout: VGPR v holds (M = v + 8*hi, N = l16 + 16*t)
    const int rbase = blockIdx.x * 128 + wave * 16 + (hi ? 8 : 0);
#pragma unroll
    for (int t = 0; t < 4; ++t) {
        const int col = l16 + t * 16;
        const float bc = bias[col];
#pragma unroll
        for (int v = 0; v < 8; ++v) {
            const int r = rbase + v;
            if (r < n) {
                const float h = acc[t][v];
                H[(size_t)r * 64 + col] = h;
                const float di = dinv[r];
                ACC[(size_t)r * 64 + col] = fmaf(h, di * di, bc);
            }
        }
    }
}

// ---------------------------------------------------------------------------
// Edge aggregation: ACC[dst] += H[src] * dinv[src]*dinv[dst]
// 16 lanes per edge, float4 per lane (full 64-wide row per 16-lane group).
// Feature buffers (25.6 MB) are L2-resident (192 MB L2) -> L2-bound atomics.
// ---------------------------------------------------------------------------
__global__ __launch_bounds__(256) void k_edge_agg(
    const long long* __restrict__ src, const long long* __restrict__ dst,
    const float* __restrict__ dinv, const float* __restrict__ H,
    float* __restrict__ ACC, int nE)
{
    const long long tid = (long long)blockIdx.x * blockDim.x + threadIdx.x;
    const int e = (int)(tid >> 4);
    const int l = (int)(tid & 15);
    if (e >= nE) return;

    const int s = (int)src[e];
    const int d = (int)dst[e];
    const float norm = dinv[s] * dinv[d];

    const float4 hv = *(const float4*)(H + (size_t)s * 64 + l * 4);
    float* p = ACC + (size_t)d * 64 + l * 4;
    atomicAdd(p + 0, hv.x * norm);
    atomicAdd(p + 1, hv.y * norm);
    atomicAdd(p + 2, hv.z * norm);
    atomicAdd(p + 3, hv.w * norm);
}

// ---------------------------------------------------------------------------
// Attention scores: s[i] = tanh(z_i @ Wa^T + ba) . q + battn ; block max.
// Wa (16 KB), ba, q staged in LDS; one node per thread.
// ---------------------------------------------------------------------------
__global__ __launch_bounds__(256) void k_attn_scores(
    const float* __restrict__ Z, const float* __restrict__ Wa,
    const float* __restrict__ ba, const float* __restrict__ q,
    const float* __restrict__ battn, float* __restrict__ S,
    float* __restrict__ Pmax, int n)
{
    __shared__ __align__(16) float lwa[64 * 64];
    __shared__ float lq[64];
    __shared__ float lb[64];
    __shared__ float red[256];

    for (int i = threadIdx.x * 4; i < 64 * 64; i += 256 * 4)
        stage_f4(Wa + i, lwa + i);
    if (threadIdx.x < 64) { lq[threadIdx.x] = q[threadIdx.x]; lb[threadIdx.x] = ba[threadIdx.x]; }
    stage_wait();
    __syncthreads();

    const int i = blockIdx.x * 256 + threadIdx.x;
    float sv = -__builtin_inff();
    if (i < n) {
        float zr[64];
#pragma unroll
        for (int k = 0; k < 64; ++k) zr[k] = Z[(size_t)i * 64 + k];
        float acc = battn[0];
        for (int j = 0; j < 64; ++j) {
            float t = lb[j];
#pragma unroll
            for (int k = 0; k < 64; ++k) t = fmaf(zr[k], lwa[j * 64 + k], t);
            acc = fmaf(tanhf(t), lq[j], acc);
        }
        S[i] = acc;
        sv = acc;
    }
    red[threadIdx.x] = sv;
    __syncthreads();
    for (int off = 128; off > 0; off >>= 1) {
        if (threadIdx.x < off)
            red[threadIdx.x] = fmaxf(red[threadIdx.x], red[threadIdx.x + off]);
        __syncthreads();
    }
    if (threadIdx.x == 0) Pmax[blockIdx.x] = red[0];
}

// Single block: reduce per-block maxima; zero softmax accumulators.
__global__ __launch_bounds__(256) void k_reduce_max_init(
    const float* __restrict__ Pmax, int nb, float* __restrict__ smax,
    float* __restrict__ numer, float* __restrict__ denom)
{
    __shared__ float red[256];
    float m = -__builtin_inff();
    for (int i = threadIdx.x; i < nb; i += 256) m = fmaxf(m, Pmax[i]);
    red[threadIdx.x] = m;
    __syncthreads();
    for (int off = 128; off > 0; off >>= 1) {
        if (threadIdx.x < off)
            red[threadIdx.x] = fmaxf(red[threadIdx.x], red[threadIdx.x + off]);
        __syncthreads();
    }
    if (threadIdx.x == 0) { smax[0] = red[0]; denom[0] = 0.0f; }
    if (threadIdx.x < 64) numer[threadIdx.x] = 0.0f;
}

// ---------------------------------------------------------------------------
// exp(s - max); accumulate denom and numer = sum_i w_i * z_i.
// Phase 1: one node/thread -> w_i in LDS + block sum of w (denominator).
// Phase 2: threads re-mapped to (chunk c=0..3, column j=0..63): lane-adjacent
// threads read Z[node*64 + j] coalesced, accumulate column partials in
// registers, reduce 4 chunks in LDS, then 64 global atomics per block.
// ---------------------------------------------------------------------------
__global__ __launch_bounds__(256) void k_attn_wsum(
    const float* __restrict__ Z, const float* __restrict__ S,
    const float* __restrict__ smax, float* __restrict__ numer,
    float* __restrict__ denom, int n)
{
    __shared__ float wbuf[256];
    __shared__ float red[256];
    __shared__ float part[4][64];

    const int base = blockIdx.x * 256;
    const int i = base + threadIdx.x;
    float w = 0.0f;
    if (i < n) w = __expf(S[i] - smax[0]);
    wbuf[threadIdx.x] = w;
    red[threadIdx.x]  = w;
    __syncthreads();
    for (int off = 128; off > 0; off >>= 1) {
        if (threadIdx.x < off) red[threadIdx.x] += red[threadIdx.x + off];
        __syncthreads();
    }
    if (threadIdx.x == 0) atomicAdd(denom, red[0]);

    const int j = threadIdx.x & 63;   // column
    const int c = threadIdx.x >> 6;   // chunk of 64 nodes
    float acc = 0.0f;
#pragma unroll 4
    for (int t = 0; t < 64; ++t) {
        const int node = base + c * 64 + t;
        const int nodec = node < n ? node : 0;        // clamp; w==0 for OOB
        acc = fmaf(wbuf[c * 64 + t], Z[(size_t)nodec * 64 + j], acc);
    }
    part[c][j] = acc;
    __syncthreads();
    if (threadIdx.x < 64) {
        const float s4 = part[0][threadIdx.x] + part[1][threadIdx.x] +
                         part[2][threadIdx.x] + part[3][threadIdx.x];
        atomicAdd(&numer[threadIdx.x], s4);
    }
}

__global__ void k_attn_final(const float* __restrict__ numer,
                             const float* __restrict__ denom,
                             float* __restrict__ out)
{
    if (threadIdx.x < 64) out[threadIdx.x] = numer[threadIdx.x] / denom[0];
}

// ---------------------------------------------------------------------------
// Launcher
// ---------------------------------------------------------------------------
extern "C" void kernel_launch(void* const* d_in, const int* in_sizes, int n_in,
                              void* d_out, int out_size, void* d_ws, size_t ws_size,
                              hipStream_t stream)
{
    const float*     x     = (const float*)d_in[0];
    const long long* ei    = (const long long*)d_in[1];
    const float*     W1    = (const float*)d_in[2];
    const float*     b1    = (const float*)d_in[3];
    const float*     W2    = (const float*)d_in[4];
    const float*     b2    = (const float*)d_in[5];
    const float*     W3    = (const float*)d_in[6];
    const float*     b3    = (const float*)d_in[7];
    const float*     Wa    = (const float*)d_in[8];
    const float*     ba    = (const float*)d_in[9];
    const float*     q     = (const float*)d_in[10];
    const float*     battn = (const float*)d_in[11];

    const int N = in_sizes[0] / 128;   // x: [N,128]
    const int E = in_sizes[1] / 2;     // edge_index: [2,E]
    const long long* src = ei;
    const long long* dst = ei + E;

    float* outz = (float*)d_out;                 // z_context [N,64]
    float* outg = outz + (size_t)N * 64;         // global_context [64]

    // Workspace layout (floats)
    float* ws = (float*)d_ws;
    size_t o = 0;
    float* deg  = ws + o; o += (size_t)N;
    float* dinv = ws + o; o += (size_t)N;
    float* bufA = ws + o; o += (size_t)N * 64;   // H (gemm output, pre-norm)
    float* bufB = ws + o; o += (size_t)N * 64;   // layer-1 accumulator
    float* bufC = ws + o; o += (size_t)N * 64;   // layer-2 accumulator
    float* S    = ws + o; o += (size_t)N;
    float* Pmax = ws + o; o += 4096;
    float* numer= ws + o; o += 64;
    float* denom= ws + o; o += 1;
    float* smax = ws + o; o += 1;
    (void)ws_size; (void)n_in; (void)out_size;

    const int T = 256;
    const int gN   = (N + T - 1) / T;
    const int gE   = (E + T - 1) / T;
    const int gNE  = (N * 64 + T - 1) / T;
    const int gRow = (N + 127) / 128;
    const long long edgeThreads = (long long)E * 16;
    const int gEd  = (int)((edgeThreads + T - 1) / T);

    // Degrees and normalization
    k_zero<<<gN, T, 0, stream>>>(deg, N);
    k_degree<<<gE, T, 0, stream>>>(dst, deg, E);
    k_dinv<<<gN, T, 0, stream>>>(deg, dinv, N);

    // Layer 1: x[N,128] @ W1 -> relu(gcn)
    k_gemm_node<128><<<gRow, T, 0, stream>>>(x, W1, b1, dinv, bufA, bufB, N);
    k_edge_agg<<<gEd, T, 0, stream>>>(src, dst, dinv, bufA, bufB, E);
    k_relu<<<gNE, T, 0, stream>>>(bufB, N * 64);

    // Layer 2
    k_gemm_node<64><<<gRow, T, 0, stream>>>(bufB, W2, b2, dinv, bufA, bufC, N);
    k_edge_agg<<<gEd, T, 0, stream>>>(src, dst, dinv, bufA, bufC, E);
    k_relu<<<gNE, T, 0, stream>>>(bufC, N * 64);

    // Layer 3: accumulator written straight into d_out (z_context)
    k_gemm_node<64><<<gRow, T, 0, stream>>>(bufC, W3, b3, dinv, bufA, outz, N);
    k_edge_agg<<<gEd, T, 0, stream>>>(src, dst, dinv, bufA, outz, E);

    // Attention pooling over nodes
    const int nb = gN;
    k_attn_scores<<<nb, T, 0, stream>>>(outz, Wa, ba, q, battn, S, Pmax, N);
    k_reduce_max_init<<<1, T, 0, stream>>>(Pmax, nb, smax, numer, denom);
    k_attn_wsum<<<nb, T, 0, stream>>>(outz, S, smax, numer, denom, N);
    k_attn_final<<<1, 64, 0, stream>>>(numer, denom, outg);
}